// SimpleTextDiffusion_69252052680881
// MI455X (gfx1250) — compile-verified
//
#include <hip/hip_runtime.h>
#include <hip/hip_bf16.h>
#include <stdint.h>

// ---------------------------------------------------------------------------
// SimpleTextDiffusion forward: bernoulli-mask tokens with per-row p=betas[t],
// labels = token where noisy==MASK else -100. Bit-exact JAX threefry2x32 RNG
// (key(1)), split-halves layout: block (i, i+n/2) -> (u[i], u[i+n/2]).
// Streaming, memory-bound: b128 NT loads/stores, 8 elements per thread,
// 4 threefry blocks per thread (each block feeds 2 elements).
// ---------------------------------------------------------------------------

typedef int   v4i __attribute__((ext_vector_type(4)));
typedef float v4f __attribute__((ext_vector_type(4)));

#define MASK_TOKEN_ID 4
#define IGNORE_INDEX  (-100)

__device__ __forceinline__ uint32_t rotl32(uint32_t x, uint32_t r) {
  // lowers to v_alignbit_b32
  return (x << r) | (x >> (32u - r));
}

// Threefry-2x32, 20 rounds, key = (0, 1)  [jax.random.key(1)]
__device__ __forceinline__ void threefry2x32_key1(uint32_t x0, uint32_t x1,
                                                  uint32_t& o0, uint32_t& o1) {
  const uint32_t ks0 = 0u;
  const uint32_t ks1 = 1u;
  const uint32_t ks2 = 0x1BD11BDAu ^ ks0 ^ ks1; // 0x1BD11BDB

  x0 += ks0; x1 += ks1;
#define TF_R4(a, b, c, d)                                \
  x0 += x1; x1 = rotl32(x1, a); x1 ^= x0;                \
  x0 += x1; x1 = rotl32(x1, b); x1 ^= x0;                \
  x0 += x1; x1 = rotl32(x1, c); x1 ^= x0;                \
  x0 += x1; x1 = rotl32(x1, d); x1 ^= x0;
  TF_R4(13, 15, 26, 6)   x0 += ks1; x1 += ks2 + 1u;
  TF_R4(17, 29, 16, 24)  x0 += ks2; x1 += ks0 + 2u;
  TF_R4(13, 15, 26, 6)   x0 += ks0; x1 += ks1 + 3u;
  TF_R4(17, 29, 16, 24)  x0 += ks1; x1 += ks2 + 4u;
  TF_R4(13, 15, 26, 6)   x0 += ks2; x1 += ks0 + 5u;
#undef TF_R4
  o0 = x0; o1 = x1;
}

// JAX uniform [0,1): bitcast((bits>>9) | 0x3F800000) - 1.0
__device__ __forceinline__ float bits_to_unit_float(uint32_t b) {
  return __uint_as_float((b >> 9) | 0x3F800000u) - 1.0f;
}

// Tiny precompute: p_row[r] = betas[t[r]]  (kills the dependent-gather chain
// in the hot kernel; p_row is 4KB, lives in L2/L1 for the main pass)
__global__ void build_prow_kernel(const int* __restrict__ t,
                                  const float* __restrict__ betas,
                                  float* __restrict__ prow, int B) {
  int i = blockIdx.x * blockDim.x + threadIdx.x;
  if (i < B) prow[i] = betas[t[i]];
}

__global__ __launch_bounds__(256)
void diffusion_kernel(const int*  __restrict__ tokens,
                      const float* __restrict__ prow,
                      float* __restrict__ out,   // [0,n): noisy, [n,2n): labels
                      int h,                      // n/2
                      int log2S,                  // log2(seq len)
                      int quarter)                // h/4 (threads)
{
  int idx = blockIdx.x * blockDim.x + threadIdx.x;
  if (idx >= quarter) return;
  int base = idx << 2;  // first lower-half element handled by this thread

  // 128-bit non-temporal token loads (read-once streaming data)
  const v4i* tok4 = (const v4i*)tokens;
  v4i tl = __builtin_nontemporal_load(tok4 + idx);            // elems base..base+3
  v4i th = __builtin_nontemporal_load(tok4 + idx + quarter);  // elems base+h..base+h+3

  float p_lo = prow[base >> log2S];
  float p_hi = prow[(base + h) >> log2S];

  v4f nz_lo, nz_hi, lb_lo, lb_hi;
#pragma unroll
  for (int c = 0; c < 4; ++c) {
    uint32_t o0, o1;
    // counter pair (i, i+h): word0 -> u[i], word1 -> u[i+h]
    threefry2x32_key1((uint32_t)(base + c), (uint32_t)(base + h + c), o0, o1);
    float u0 = bits_to_unit_float(o0);
    float u1 = bits_to_unit_float(o1);

    int t0 = tl[c], t1 = th[c];
    int n0 = (u0 < p_lo) ? MASK_TOKEN_ID : t0;
    int n1 = (u1 < p_hi) ? MASK_TOKEN_ID : t1;
    // faithful to reference: label kept if noisy==MASK (covers pre-existing 4s)
    int l0 = (n0 == MASK_TOKEN_ID) ? t0 : IGNORE_INDEX;
    int l1 = (n1 == MASK_TOKEN_ID) ? t1 : IGNORE_INDEX;

    nz_lo[c] = (float)n0; nz_hi[c] = (float)n1;
    lb_lo[c] = (float)l0; lb_hi[c] = (float)l1;
  }

  // 128-bit non-temporal stores (write-once, never re-read -> don't rinse L2)
  v4f* outn = (v4f*)out;
  v4f* outl = (v4f*)(out + (size_t)h * 2);  // labels start at element n = 2h
  __builtin_nontemporal_store(nz_lo, outn + idx);
  __builtin_nontemporal_store(nz_hi, outn + idx + quarter);
  __builtin_nontemporal_store(lb_lo, outl + idx);
  __builtin_nontemporal_store(lb_hi, outl + idx + quarter);
}

extern "C" void kernel_launch(void* const* d_in, const int* in_sizes, int n_in,
                              void* d_out, int out_size, void* d_ws, size_t ws_size,
                              hipStream_t stream) {
  const int*   tokens = (const int*)d_in[0];
  const int*   t      = (const int*)d_in[1];
  const float* betas  = (const float*)d_in[2];
  float*       out    = (float*)d_out;

  const int n = in_sizes[0];          // B*S = 8388608 (even, pow2)
  const int B = in_sizes[1];          // 1024
  const int S = n / B;                // 8192
  const int log2S = 31 - __builtin_clz((unsigned)S);
  const int h = n >> 1;               // split-halves midpoint
  const int quarter = h >> 2;         // threads (4 lower + 4 upper elems each)

  float* prow = (float*)d_ws;         // 4KB scratch: per-row probability

  build_prow_kernel<<<(B + 255) / 256, 256, 0, stream>>>(t, betas, prow, B);
  diffusion_kernel<<<(quarter + 255) / 256, 256, 0, stream>>>(
      tokens, prow, out, h, log2S, quarter);
}